// DistSageConv_82197084110915
// MI455X (gfx1250) — compile-verified
//
#include <hip/hip_runtime.h>
#include <stdint.h>

#define FEAT 128
#define LDA 132  // padded LDS row stride (floats): banks = (4*row + k) % 64 -> conflict-free

typedef __attribute__((ext_vector_type(2))) float v2f;
typedef __attribute__((ext_vector_type(4))) float f32x4;
typedef __attribute__((ext_vector_type(8))) float v8f;

static constexpr int N_DST  = 100000;
static constexpr int N_EDGE = 500000;
static constexpr int TILES  = N_DST / 16;   // 6250
static constexpr int GBLK   = 1250;         // persistent workgroups
static constexpr int TPB    = TILES / GBLK; // 5 tiles per workgroup

// ---- CDNA5 async global->LDS helpers (ASYNCcnt-tracked DMA, per-lane) ----
__device__ __forceinline__ uint32_t lds_off(const void* p) {
    // generic pointer to __shared__: low 32 bits == LDS byte offset
    return (uint32_t)(uint64_t)p;
}
__device__ __forceinline__ void async_copy_b128(uint32_t dst_lds, const void* src) {
    asm volatile("global_load_async_to_lds_b128 %0, %1, off"
                 :: "v"(dst_lds), "v"(src) : "memory");
}
__device__ __forceinline__ void wait_async0() {
    asm volatile("s_wait_asynccnt 0x0" ::: "memory");
}

// ---------------- Kernel 1: zero the aggregation scratch ----------------
__global__ void zero_kernel(f32x4* __restrict__ p, int n4) {
    int i = blockIdx.x * blockDim.x + threadIdx.x;
    if (i < n4) p[i] = f32x4{0.f, 0.f, 0.f, 0.f};
}

// ---------------- Kernel 2: edge gather + scatter-add -------------------
// One wave32 per edge; each lane moves 4 contiguous floats (b128 gather,
// 4x native global_atomic_add_f32). x table (102 MB) stays L2-resident.
__global__ __launch_bounds__(256) void scatter_kernel(
    const float* __restrict__ x,
    const int* __restrict__ src_idx,
    const int* __restrict__ dst_idx,
    float* __restrict__ agg) {
    int e = blockIdx.x * 8 + (threadIdx.x >> 5);
    if (e >= N_EDGE) return;
    int lane = threadIdx.x & 31;
    const int s = src_idx[e];
    const int d = dst_idx[e];
    f32x4 v = ((const f32x4*)(x + (size_t)s * FEAT))[lane];
    float* base = agg + (size_t)d * FEAT + lane * 4;
    __hip_atomic_fetch_add(base + 0, v.x, __ATOMIC_RELAXED, __HIP_MEMORY_SCOPE_AGENT);
    __hip_atomic_fetch_add(base + 1, v.y, __ATOMIC_RELAXED, __HIP_MEMORY_SCOPE_AGENT);
    __hip_atomic_fetch_add(base + 2, v.z, __ATOMIC_RELAXED, __HIP_MEMORY_SCOPE_AGENT);
    __hip_atomic_fetch_add(base + 3, v.w, __ATOMIC_RELAXED, __HIP_MEMORY_SCOPE_AGENT);
}

// ---------------- Kernel 3: persistent fused dual GEMM via f32 WMMA -----
// Each block stages W1/W2 once, then pipelines TPB row-tiles:
//   wait_asynccnt -> barrier -> issue async A(t+1) -> 64x v_wmma on A(t).
// Degree normalization is applied to the accumulator AFTER the first WMMA
// chain (it commutes: per-row scalar), so A staging is a pure async copy.
__global__ __launch_bounds__(256) void gemm_kernel(
    const float* __restrict__ x,
    const float* __restrict__ W1,
    const float* __restrict__ W2,
    const float* __restrict__ degree,
    const int* __restrict__ self_ids,
    const float* __restrict__ agg,
    float* __restrict__ out) {
    __shared__ __align__(16) float sW1[FEAT][LDA];
    __shared__ __align__(16) float sW2[FEAT][LDA];
    __shared__ __align__(16) float sA1[2][16][LDA];  // raw agg tile (double-buffered)
    __shared__ __align__(16) float sA2[2][16][LDA];  // self-gathered x tile
    __shared__ float sInvDeg[2][16];

    const int tid = threadIdx.x;

    // One-time async staging of W1/W2 (4096 b128 transfers each).
    for (int i = tid; i < FEAT * FEAT / 4; i += 256) {
        int r  = i >> 5;
        int c4 = i & 31;
        async_copy_b128(lds_off(&sW1[r][c4 * 4]), W1 + i * 4);
        async_copy_b128(lds_off(&sW2[r][c4 * 4]), W2 + i * 4);
    }

    const int tile0 = blockIdx.x * TPB;

    auto stage_A = [&](int buf, int tile) {
        int row0 = tile * 16;
        for (int i = tid; i < 16 * 32; i += 256) {  // 2 iters/thread, 4 async issues
            int r  = i >> 5;
            int c4 = i & 31;
            int g  = row0 + r;
            async_copy_b128(lds_off(&sA1[buf][r][c4 * 4]),
                            agg + (size_t)g * FEAT + c4 * 4);
            int sid = self_ids[g];
            async_copy_b128(lds_off(&sA2[buf][r][c4 * 4]),
                            x + (size_t)sid * FEAT + c4 * 4);
        }
        if (tid < 16) sInvDeg[buf][tid] = 1.0f / degree[row0 + tid];
    };

    stage_A(0, tile0);  // prologue

    const int wave  = tid >> 5;          // n-tile: output cols [16*wave, 16*wave+16)
    const int lane  = tid & 31;
    const int nb    = wave * 16;
    const int hrow  = lane & 15;         // A row / W row within tile
    const int koff  = (lane >> 4) * 2;   // lanes 0-15: K+0..1, lanes 16-31: K+2..3
    const int mbase = (lane >> 4) * 8;   // D VGPR r -> output row mbase + r

    for (int j = 0; j < TPB; ++j) {
        const int tile = tile0 + j;
        const int p    = j & 1;

        wait_async0();        // this wave's async copies for tile j landed
        __syncthreads();      // all waves' copies landed (incl. W on j==0)

        if (j + 1 < TPB) stage_A(1 - p, tile + 1);  // overlap copy with WMMAs

        v8f c = {};
        #pragma unroll
        for (int kk = 0; kk < FEAT; kk += 4) {
            v2f a = *(const v2f*)&sA1[p][hrow][kk + koff];
            v2f b = *(const v2f*)&sW1[nb + hrow][kk + koff];
            c = __builtin_amdgcn_wmma_f32_16x16x4_f32(
                    false, a, false, b, (short)0, c, false, false);
        }
        // per-output-row degree scale (commutes with the row-linear GEMM)
        #pragma unroll
        for (int r = 0; r < 8; ++r) c[r] *= sInvDeg[p][mbase + r];
        #pragma unroll
        for (int kk = 0; kk < FEAT; kk += 4) {
            v2f a = *(const v2f*)&sA2[p][hrow][kk + koff];
            v2f b = *(const v2f*)&sW2[nb + hrow][kk + koff];
            c = __builtin_amdgcn_wmma_f32_16x16x4_f32(
                    false, a, false, b, (short)0, c, false, false);
        }

        const int row0 = tile * 16;
        const int ocol = nb + hrow;
        #pragma unroll
        for (int r = 0; r < 8; ++r) {
            out[(size_t)(row0 + mbase + r) * FEAT + ocol] = c[r];
        }
    }
}

extern "C" void kernel_launch(void* const* d_in, const int* in_sizes, int n_in,
                              void* d_out, int out_size, void* d_ws, size_t ws_size,
                              hipStream_t stream) {
    const float* x        = (const float*)d_in[0];
    const float* W1       = (const float*)d_in[1];
    const float* W2       = (const float*)d_in[2];
    const float* degree   = (const float*)d_in[3];
    const int*   src_idx  = (const int*)d_in[4];
    const int*   dst_idx  = (const int*)d_in[5];
    const int*   self_ids = (const int*)d_in[6];
    float* out = (float*)d_out;
    float* agg = (float*)d_ws;  // N_DST * 128 floats = 51.2 MB scratch

    const int n4 = N_DST * FEAT / 4;
    zero_kernel<<<(n4 + 255) / 256, 256, 0, stream>>>((f32x4*)agg, n4);
    scatter_kernel<<<N_EDGE / 8, 256, 0, stream>>>(x, src_idx, dst_idx, agg);
    gemm_kernel<<<GBLK, 256, 0, stream>>>(x, W1, W2, degree, self_ids, agg, out);
}